// BidPrefix_1829656068135
// MI455X (gfx1250) — compile-verified
//
#include <hip/hip_runtime.h>

// BidPrefix: per-row cumprod gather.
//   survival = cp[bid],  rate_last = cp[mp] - cp[mp+1],  cp[k] = prod_{i<k} probs[i]
// One wave32 per row; lane L owns probs[16L..16L+15].
// Memory-bound: read only elements < max(bid, mp+1) with NT hints.

#define SEQLEN 512
#define ROWSTRIDE 514          // 512 probs + mp + bid
#define WAVES_PER_BLOCK 8
#define LDS_ROW (SEQLEN + 8)   // 520 floats; 2080 B per wave slice (16B-aligned)

typedef float v2f __attribute__((ext_vector_type(2)));   // native vec for NT builtin

__global__ __launch_bounds__(256) void bidprefix_kernel(const float* __restrict__ in,
                                                        float* __restrict__ out,
                                                        int B) {
    __shared__ float cp_lds[WAVES_PER_BLOCK][LDS_ROW];

    const int lane = threadIdx.x & 31;
    const int wave = threadIdx.x >> 5;
    const int row  = blockIdx.x * WAVES_PER_BLOCK + wave;
    if (row >= B) return;                       // uniform per wave

    const float* rp = in + (size_t)row * ROWSTRIDE;

    // ---- header (wave-uniform): mp, bid. 8B-aligned float2 load. ----
    float2 hdr = *(const float2*)(rp + SEQLEN);
    const int mp   = (int)hdr.x;                // [0, 512)
    const int bid  = (int)hdr.y;                // [0, 512]
    const int tmax = max(bid, mp + 1);          // highest cp index needed

    // ---- bulk load: lane L owns [16L, 16L+16); skip if segment >= tmax ----
    const int base = lane << 4;
    float v[16];
#pragma unroll
    for (int j = 0; j < 16; ++j) v[j] = 1.0f;
    if (base < tmax) {
        const v2f* p2 = (const v2f*)(rp + base);         // 8B-aligned
#pragma unroll
        for (int j = 0; j < 8; ++j) {
            v2f t = __builtin_nontemporal_load(p2 + j);  // read-once: TH=NT
            v[2 * j]     = t.x;
            v[2 * j + 1] = t.y;
        }
    }

    // ---- per-lane inclusive prefix products ----
    float r[16];
    r[0] = v[0];
#pragma unroll
    for (int j = 1; j < 16; ++j) r[j] = r[j - 1] * v[j];

    // ---- wave-exclusive product scan of full lane products (wave32) ----
    float x = r[15];
#pragma unroll
    for (int o = 1; o < 32; o <<= 1) {
        float y = __shfl_up(x, o, 32);
        if (lane >= o) x *= y;
    }
    float S = __shfl_up(x, 1, 32);              // exclusive scan
    if (lane == 0) S = 1.0f;                    // cp[0] = 1

    // ---- spill cp[16L .. 16L+15] to this wave's LDS slice (64B chunk/lane) ----
    float* wslot = &cp_lds[wave][base];         // 16B-aligned -> ds_store_b128
    wslot[0] = S;
#pragma unroll
    for (int j = 1; j < 16; ++j) wslot[j] = S * r[j - 1];
    if (lane == 31) cp_lds[wave][SEQLEN] = S * r[15];   // cp[512]

    // DS ops from the same wave are in-order: stores above precede loads below.

    // ---- dynamic-index gather + output (lane 0) ----
    if (lane == 0) {
        float surv = cp_lds[wave][bid];
        float rl   = cp_lds[wave][mp] - cp_lds[wave][mp + 1];
        out[row]     = surv;   // survival_rate, flat (B)
        out[B + row] = rl;     // rate_last, flat (B)
    }
}

extern "C" void kernel_launch(void* const* d_in, const int* in_sizes, int n_in,
                              void* d_out, int out_size, void* d_ws, size_t ws_size,
                              hipStream_t stream) {
    const float* in  = (const float*)d_in[0];
    float*       out = (float*)d_out;
    const int B = in_sizes[0] / ROWSTRIDE;      // 65536

    dim3 block(256);
    dim3 grid((B + WAVES_PER_BLOCK - 1) / WAVES_PER_BLOCK);
    hipLaunchKernelGGL(bidprefix_kernel, grid, block, 0, stream, in, out, B);
}